// UpBlock_60361470378567
// MI455X (gfx1250) — compile-verified
//
#include <hip/hip_runtime.h>
#include <math.h>

// ---------------------------------------------------------------------------
// MI455X (gfx1250) fused UpBlock + Mamba pipeline, fp32 WMMA (16x16x4) path.
// Round 2: TDM (tensor_load_to_lds) staging + LDS halo tiles + register
// blocking to cut VMEM-per-WMMA; packed weight layouts for contiguous loads.
// ---------------------------------------------------------------------------

typedef __attribute__((ext_vector_type(2))) float v2f;
typedef __attribute__((ext_vector_type(4))) float v4f;
typedef __attribute__((ext_vector_type(8))) float v8f;
typedef __attribute__((ext_vector_type(4))) unsigned int v4u;
typedef __attribute__((ext_vector_type(4))) int v4i;
typedef __attribute__((ext_vector_type(8))) int v8i;

#if defined(__has_builtin)
#if __has_builtin(__builtin_amdgcn_tensor_load_to_lds)
#define HAVE_TDM 1
#endif
#endif

#define DEV __device__ __forceinline__

DEV v8f wmma4(v2f a, v2f b, v8f c) {
  // V_WMMA_F32_16X16X4_F32: D = A(16x4) * B(4x16) + C(16x16), fp32 throughout.
  return __builtin_amdgcn_wmma_f32_16x16x4_f32(false, a, false, b, (short)0, c,
                                               false, false);
}

DEV float siluf(float x) { return x / (1.f + __expf(-x)); }

static constexpr int BATCH = 8;
static constexpr int HW = 4096;
static constexpr int CIN = 256;
static constexpr int DI = 256;   // d_inner
static constexpr int DBLP = 48;  // dt_rank(8) + 2*d_state(32) = 40, padded to 48
static constexpr long ROWS = (long)BATCH * HW;  // 32768

// workspace layout (float offsets)
static constexpr size_t OFF_XCAT = 0;                        // 8*4096*256
static constexpr size_t SZ_XCAT  = (size_t)BATCH * HW * 256;
static constexpr size_t OFF_SEQ  = OFF_XCAT + SZ_XCAT;       // 32768*128
static constexpr size_t SZ_SEQ   = (size_t)ROWS * 128;
static constexpr size_t OFF_XZ   = OFF_SEQ + SZ_SEQ;         // 32768*512
static constexpr size_t SZ_XZ    = (size_t)ROWS * 2 * DI;
static constexpr size_t OFF_XC   = OFF_XZ + SZ_XZ;           // 32768*256
static constexpr size_t SZ_XC    = (size_t)ROWS * DI;
static constexpr size_t OFF_DBL  = OFF_XC + SZ_XC;           // 32768*48
static constexpr size_t SZ_DBL   = (size_t)ROWS * DBLP;
static constexpr size_t OFF_DT   = OFF_DBL + SZ_DBL;         // 32768*256
static constexpr size_t SZ_DT    = (size_t)ROWS * DI;
static constexpr size_t OFF_STAT = OFF_DT + SZ_DT;           // 256 floats
static constexpr size_t OFF_X1CL = OFF_STAT + 256;           // x1 channel-last
static constexpr size_t SZ_X1CL  = (size_t)BATCH * 1024 * 256;
static constexpr size_t OFF_WTK  = OFF_X1CL + SZ_X1CL;       // Wt packed [ij][n][k]
static constexpr size_t SZ_WTK   = (size_t)4 * 128 * 256;
static constexpr size_t OFF_WCK  = OFF_WTK + SZ_WTK;         // Wc packed [tap][n][k]
static constexpr size_t SZ_WCK   = (size_t)9 * 128 * 256;
static constexpr size_t OFF_YSSM = OFF_XCAT;  // reuse xcat (dead after conv3x3)

// ---------------------------------------------------------------------------
// pack kernels: make every WMMA operand contiguous along K
__global__ void k_pack_x2(const float* __restrict__ x2, float* __restrict__ xcat) {
  size_t idx = (size_t)blockIdx.x * blockDim.x + threadIdx.x;
  if (idx >= (size_t)BATCH * HW * 128) return;
  int c = (int)(idx & 127);
  size_t bp = idx >> 7;            // b*4096 + pos
  size_t b = bp >> 12;
  size_t pos = bp & 4095;
  xcat[bp * 256 + c] = x2[(b * 128 + c) * 4096 + pos];
}

__global__ void k_pack_x1(const float* __restrict__ x1, float* __restrict__ x1cl) {
  size_t idx = (size_t)blockIdx.x * blockDim.x + threadIdx.x;
  if (idx >= (size_t)BATCH * 1024 * 256) return;
  int k = (int)(idx & 255);
  size_t bp = idx >> 8;            // b*1024 + pos
  size_t b = bp >> 10;
  size_t pos = bp & 1023;
  x1cl[bp * 256 + k] = x1[(b * 256 + k) * 1024 + pos];
}

__global__ void k_pack_wt(const float* __restrict__ wt, float* __restrict__ wtk) {
  int idx = blockIdx.x * blockDim.x + threadIdx.x;  // 4*128*256
  if (idx >= 4 * 128 * 256) return;
  int k = idx & 255;
  int n = (idx >> 8) & 127;
  int ij = idx >> 15;
  wtk[idx] = wt[(size_t)k * 512 + n * 4 + ij];      // Wt[k,n,i,j]
}

__global__ void k_pack_wc(const float* __restrict__ wc, float* __restrict__ wck) {
  int idx = blockIdx.x * blockDim.x + threadIdx.x;  // 9*128*256
  if (idx >= 9 * 128 * 256) return;
  int k = idx & 255;
  int n = (idx >> 8) & 127;
  int tap = idx >> 15;
  wck[idx] = wc[(size_t)n * 2304 + k * 9 + tap];    // Wc[n,k,dy,dx]
}

// ---------------------------------------------------------------------------
// 1. conv_transpose 2x2 stride 2 as 4 sub-pixel GEMMs; M=8192, N=128, K=256.
__global__ void __launch_bounds__(32) k_upsample_wmma(
    const float* __restrict__ x1cl, const float* __restrict__ wtk,
    const float* __restrict__ btb, float* __restrict__ xcat) {
  int lane = threadIdx.x;
  int m = lane & 15, kg = lane >> 4;
  int tM = blockIdx.x;             // 512 tiles of 16 rows over B*H1*W1
  int tN = blockIdx.y;             // 8 tiles of 16 out channels
  int ij = blockIdx.z;
  int i = ij >> 1, j = ij & 1;
  int rA = tM * 16 + m;
  int n = tN * 16 + m;
  const float* ab = x1cl + (size_t)rA * 256 + kg * 2;
  const float* bb = wtk + ((size_t)ij * 128 + n) * 256 + kg * 2;
  v8f acc = {};
  for (int ks = 0; ks < 64; ++ks) {
    v2f a = *(const v2f*)(ab + ks * 4);
    v2f b = *(const v2f*)(bb + ks * 4);
    acc = wmma4(a, b, acc);
  }
  float bias = btb[n];
#pragma unroll
  for (int v = 0; v < 8; ++v) {
    int r = tM * 16 + v + kg * 8;
    int b2 = r >> 10, p = r & 1023;
    int h1 = p >> 5, w1 = p & 31;
    int h = 2 * h1 + i, w = 2 * w1 + j;
    xcat[((size_t)b2 * 4096 + h * 64 + w) * 256 + 128 + n] = acc[v] + bias;
  }
}

// ---------------------------------------------------------------------------
// 2. conv3x3 SAME: LDS halo tile (3 x 18 x 256, cell stride 260 for
//    conflict-free ds_load_b64), 4 waves x 2 N-tiles sharing the A fragment.
__global__ void __launch_bounds__(128) k_conv3x3_wmma(
    const float* __restrict__ xcat, const float* __restrict__ wck,
    const float* __restrict__ bcb, float* __restrict__ y) {
  __shared__ float halo[3 * 18 * 260];
  int tid = threadIdx.x;
  int r0 = blockIdx.x * 16;        // 16 consecutive w within one image row
  int b = r0 >> 12, pos = r0 & 4095;
  int h = pos >> 6, w0 = pos & 63;
  // stage halo with zero-filled borders (3456 float4)
  for (int i = tid; i < 3456; i += 128) {
    int e = i << 2;
    int rr = e / 4608;
    int rem = e - rr * 4608;
    int col = rem >> 8, k = rem & 255;
    int hh = h - 1 + rr, ww = w0 - 1 + col;
    v4f val = {};
    if (hh >= 0 && hh < 64 && ww >= 0 && ww < 64)
      val = *(const v4f*)(xcat + (((size_t)b * 4096 + hh * 64 + ww) << 8) + k);
    *(v4f*)&halo[(rr * 18 + col) * 260 + k] = val;
  }
  __syncthreads();
  int wv = tid >> 5, lane = tid & 31;
  int m = lane & 15, kg = lane >> 4;
  int n0 = (2 * wv) * 16 + m, n1 = n0 + 16;
  v8f acc0 = {}, acc1 = {};
  for (int dy = 0; dy < 3; ++dy) {
    for (int dx = 0; dx < 3; ++dx) {
      int tap = dy * 3 + dx;
      const float* arow = halo + (dy * 18 + dx + m) * 260 + kg * 2;
      const float* wb0 = wck + ((size_t)tap * 128 + n0) * 256 + kg * 2;
      const float* wb1 = wck + ((size_t)tap * 128 + n1) * 256 + kg * 2;
      for (int ks = 0; ks < 64; ++ks) {
        v2f a = *(const v2f*)(arow + ks * 4);   // ds_load_b64, shared by 2 tiles
        v2f b0 = *(const v2f*)(wb0 + ks * 4);
        v2f b1 = *(const v2f*)(wb1 + ks * 4);
        acc0 = wmma4(a, b0, acc0);
        acc1 = wmma4(a, b1, acc1);
      }
    }
  }
  float bi0 = bcb[n0], bi1 = bcb[n1];
#pragma unroll
  for (int v = 0; v < 8; ++v) {
    size_t r = (size_t)(r0 + v + kg * 8) * 128;
    y[r + n0] = acc0[v] + bi0;
    y[r + n1] = acc1[v] + bi1;
  }
}

// ---------------------------------------------------------------------------
// 3. BN stats: per-channel sum / sumsq (coalesced: thread == channel)
__global__ void k_bnstats(const float* __restrict__ y, float* __restrict__ stats) {
  int c = threadIdx.x;             // 0..127
  size_t r0 = (size_t)blockIdx.x * 256;
  float s = 0.f, sq = 0.f;
  for (int i = 0; i < 256; ++i) {
    float v = y[(r0 + i) * 128 + c];
    s += v; sq += v * v;
  }
  atomicAdd(&stats[c], s);
  atomicAdd(&stats[128 + c], sq);
}

// 4. BN apply + ReLU in-place
__global__ void k_bnrelu(float* __restrict__ y, const float* __restrict__ stats,
                         const float* __restrict__ g, const float* __restrict__ be) {
  size_t idx = (size_t)blockIdx.x * blockDim.x + threadIdx.x;
  if (idx >= (size_t)ROWS * 128) return;
  int c = (int)(idx & 127);
  const float invN = 1.f / 32768.f;
  float mean = stats[c] * invN;
  float var = stats[128 + c] * invN - mean * mean;
  float v = y[idx];
  v = g[c] * (v - mean) * rsqrtf(var + 1e-5f) + be[c];
  y[idx] = fmaxf(v, 0.f);
}

// ---------------------------------------------------------------------------
// 5. in_proj: M=32768, N=512, K=128. TDM stages a 64x128 A-tile into LDS with
//    DMA padding (4 dwords per 128) -> 132-float row stride, conflict-free
//    ds_load_b64. 8 waves x 4 N-tiles register blocking.
__global__ void __launch_bounds__(256) k_inproj_wmma(
    const float* __restrict__ seq, const float* __restrict__ wp,
    float* __restrict__ xz) {
  __shared__ float At[64 * 132];
  int tid = threadIdx.x;
  size_t row0 = (size_t)blockIdx.x * 64;
#if defined(HAVE_TDM)
  if (tid < 32) {                  // wave 0 issues the DMA (TDM ignores EXEC)
    unsigned lds = (unsigned)(size_t)(void*)At;  // low 32 bits = LDS offset
    unsigned long long ga = (unsigned long long)(const void*)(seq + row0 * 128);
    v4u g0;
    g0.x = 1u;                                          // count = 1 (valid D#)
    g0.y = lds;                                         // lds_addr
    g0.z = (unsigned)(ga & 0xffffffffu);                // global_addr[31:0]
    g0.w = (unsigned)((ga >> 32) & 0x01ffffffu) | (2u << 30);  // addr[56:32]|type=2
    v8i g1;
    g1[0] = (int)((2u << 16) |        // data_size = 4B
                  (1u << 20) |        // pad_enable
                  (6u << 22) |        // pad_interval: every 128 dwords
                  (3u << 25));        // pad_amount: 4 dwords -> row stride 132
    g1[1] = (int)(128u << 16);        // tensor_dim0 = 128 (K)
    g1[2] = (int)(32768u << 16);      // tensor_dim1 = 32768 rows
    g1[3] = (int)(128u << 16);        // tile_dim0 = 128
    g1[4] = 64;                       // tile_dim1 = 64 rows
    g1[5] = 128;                      // tensor_dim0_stride = 128
    g1[6] = 0;
    g1[7] = 0;
    v4i g2 = {}, g3 = {};
#if __clang_major__ >= 23
    v8i g4 = {};
    __builtin_amdgcn_tensor_load_to_lds(g0, g1, g2, g3, g4, 0);
#else
    __builtin_amdgcn_tensor_load_to_lds(g0, g1, g2, g3, 0);
#endif
    __builtin_amdgcn_s_wait_tensorcnt(0);
  }
  __syncthreads();
#else
  for (int i = tid; i < 64 * 32; i += 256) {  // fallback: cooperative stage
    int r = i >> 5, q = (i & 31) << 2;
    *(v4f*)&At[r * 132 + q] = *(const v4f*)(seq + (row0 + r) * 128 + q);
  }
  __syncthreads();
#endif
  int wv = tid >> 5, lane = tid & 31;
  int m = lane & 15, kg = lane >> 4;
  for (int ms = 0; ms < 4; ++ms) {
    v8f acc[4] = {{}, {}, {}, {}};
    const float* arow = At + (ms * 16 + m) * 132 + kg * 2;
    const float* bb = wp + (size_t)((wv * 4) * 16 + m) * 128 + kg * 2;
    for (int ks = 0; ks < 32; ++ks) {
      v2f a = *(const v2f*)(arow + ks * 4);   // one ds_load feeds 4 WMMA
#pragma unroll
      for (int j = 0; j < 4; ++j) {
        v2f b = *(const v2f*)(bb + (size_t)j * 16 * 128 + ks * 4);
        acc[j] = wmma4(a, b, acc[j]);
      }
    }
#pragma unroll
    for (int j = 0; j < 4; ++j) {
      int n = (wv * 4 + j) * 16 + m;
#pragma unroll
      for (int v = 0; v < 8; ++v)
        xz[(row0 + ms * 16 + v + kg * 8) * 512 + n] = acc[j][v];
    }
  }
}

// ---------------------------------------------------------------------------
// 6. depthwise causal conv1d (k=4) + SiLU on x_in = xz[:, 0:256]
__global__ void k_conv1d_silu(const float* __restrict__ xz,
                              const float* __restrict__ cw,
                              const float* __restrict__ cb,
                              float* __restrict__ xc) {
  size_t idx = (size_t)blockIdx.x * blockDim.x + threadIdx.x;
  if (idx >= (size_t)ROWS * 256) return;
  int d = (int)(idx & 255);
  size_t row = idx >> 8;           // b*4096 + t
  int t = (int)(row & 4095);
  float acc = cb[d];
#pragma unroll
  for (int k = 0; k < 4; ++k) {
    int tt = t - 3 + k;
    if (tt >= 0) acc += cw[d * 4 + k] * xz[(row - 3 + k) * 512 + d];
  }
  xc[row * 256 + d] = siluf(acc);
}

// ---------------------------------------------------------------------------
// 7. x_proj: dbl[r, n] = sum_k xc[r, k] * Wx[n, k]; N=40 (3 tiles, masked)
__global__ void __launch_bounds__(32) k_xproj_wmma(
    const float* __restrict__ xc, const float* __restrict__ xw,
    float* __restrict__ dbl) {
  int lane = threadIdx.x, m = lane & 15, kg = lane >> 4;
  int r = blockIdx.x * 16 + m;
  int n = blockIdx.y * 16 + m;     // up to 47
  bool nok = (n < 40);
  const float* ab = xc + (size_t)r * 256 + kg * 2;
  const float* bb = xw + (size_t)n * 256 + kg * 2;
  v8f acc = {};
  for (int ks = 0; ks < 64; ++ks) {
    v2f a = *(const v2f*)(ab + ks * 4);
    v2f b;
    if (nok) b = *(const v2f*)(bb + ks * 4);
    else { b.x = 0.f; b.y = 0.f; }
    acc = wmma4(a, b, acc);
  }
  if (nok) {
#pragma unroll
    for (int v = 0; v < 8; ++v)
      dbl[(size_t)(blockIdx.x * 16 + v + kg * 8) * DBLP + n] = acc[v];
  }
}

// ---------------------------------------------------------------------------
// 8. dt_proj (K=8) + softplus
__global__ void __launch_bounds__(32) k_dtproj_wmma(
    const float* __restrict__ dbl, const float* __restrict__ dw,
    const float* __restrict__ db, float* __restrict__ dt) {
  int lane = threadIdx.x, m = lane & 15, kg = lane >> 4;
  int r = blockIdx.x * 16 + m;
  int n = blockIdx.y * 16 + m;     // 0..255
  const float* ab = dbl + (size_t)r * DBLP + kg * 2;
  const float* bb = dw + (size_t)n * 8 + kg * 2;
  v8f acc = {};
#pragma unroll
  for (int ks = 0; ks < 2; ++ks) {
    v2f a = *(const v2f*)(ab + ks * 4);
    v2f b = *(const v2f*)(bb + ks * 4);
    acc = wmma4(a, b, acc);
  }
  float bias = db[n];
#pragma unroll
  for (int v = 0; v < 8; ++v) {
    float x = acc[v] + bias;
    float sp = (x > 20.f) ? x : log1pf(__expf(x));
    dt[(size_t)(blockIdx.x * 16 + v + kg * 8) * 256 + n] = sp;
  }
}

// ---------------------------------------------------------------------------
// 9. selective scan: thread = (channel d, state n); 4096 sequential steps.
__global__ void __launch_bounds__(1024) k_scan(
    const float* __restrict__ dt, const float* __restrict__ dbl,
    const float* __restrict__ xc, const float* __restrict__ alog,
    const float* __restrict__ Dp, float* __restrict__ yssm) {
  int b = blockIdx.x;              // batch
  int dl = threadIdx.x >> 4;       // local channel 0..63
  int n = threadIdx.x & 15;        // state index
  int d = blockIdx.y * 64 + dl;
  float Acoef = -__expf(alog[d * 16 + n]);  // A = -exp(A_log)
  float Dd = Dp[d];
  float h = 0.f;
  size_t rowbase = (size_t)b * 4096;
  for (int t = 0; t < 4096; ++t) {
    size_t row = rowbase + t;
    float dtv = dt[row * 256 + d];
    float xv = xc[row * 256 + d];
    float Bn = dbl[row * DBLP + 8 + n];
    float Cn = dbl[row * DBLP + 24 + n];
    h = __expf(dtv * Acoef) * h + (dtv * xv) * Bn;
    float p = h * Cn;
    p += __shfl_xor(p, 1, 32);
    p += __shfl_xor(p, 2, 32);
    p += __shfl_xor(p, 4, 32);
    p += __shfl_xor(p, 8, 32);
    if (n == 0) yssm[row * 256 + d] = p + Dd * xv;
    if ((t & 7) == 0 && t + 8 < 4096) {  // global_prefetch_b8 ahead of the chain
      __builtin_prefetch(dt + (row + 8) * 256 + d, 0, 0);
      __builtin_prefetch(dbl + (row + 8) * DBLP + n, 0, 0);
    }
  }
}

// ---------------------------------------------------------------------------
// 10. out_proj with fused gate A = y_ssm * silu(z) and fused LayerNorm.
__global__ void __launch_bounds__(128) k_outproj_ln(
    const float* __restrict__ yssm, const float* __restrict__ xz,
    const float* __restrict__ ow, const float* __restrict__ lng,
    const float* __restrict__ lnb, float* __restrict__ out) {
  __shared__ float tile[16][130];
  __shared__ float mu[16], rs[16];
  int tid = threadIdx.x;
  int wave = tid >> 5, lane = tid & 31;
  int m = lane & 15, kg = lane >> 4;
  int rA = blockIdx.x * 16 + m;
  int n0 = (wave * 2) * 16 + m, n1 = (wave * 2 + 1) * 16 + m;
  const float* yb = yssm + (size_t)rA * 256 + kg * 2;
  const float* zb = xz + (size_t)rA * 512 + 256 + kg * 2;
  const float* b0 = ow + (size_t)n0 * 256 + kg * 2;
  const float* b1 = ow + (size_t)n1 * 256 + kg * 2;
  v8f acc0 = {}, acc1 = {};
  for (int ks = 0; ks < 64; ++ks) {
    v2f yv = *(const v2f*)(yb + ks * 4);
    v2f zv = *(const v2f*)(zb + ks * 4);
    v2f a;
    a.x = yv.x * siluf(zv.x);
    a.y = yv.y * siluf(zv.y);
    v2f w0 = *(const v2f*)(b0 + ks * 4);
    v2f w1 = *(const v2f*)(b1 + ks * 4);
    acc0 = wmma4(a, w0, acc0);
    acc1 = wmma4(a, w1, acc1);
  }
#pragma unroll
  for (int v = 0; v < 8; ++v) {
    int mr = v + kg * 8;
    tile[mr][n0] = acc0[v];
    tile[mr][n1] = acc1[v];
  }
  __syncthreads();
  if (tid < 16) {
    float s = 0.f, sq = 0.f;
    for (int c = 0; c < 128; ++c) { float vv = tile[tid][c]; s += vv; sq += vv * vv; }
    float mean = s * (1.f / 128.f);
    float var = sq * (1.f / 128.f) - mean * mean;
    mu[tid] = mean;
    rs[tid] = rsqrtf(var + 1e-5f);
  }
  __syncthreads();
  int gr0 = blockIdx.x * 16;
  int bo = gr0 >> 12;
  int pos0 = gr0 & 4095;
  int rr = tid & 15;
  int cb = tid >> 4;               // 0..7
  for (int it = 0; it < 16; ++it) {
    int c = cb + it * 8;
    float val = lng[c] * (tile[rr][c] - mu[rr]) * rs[rr] + lnb[c];
    out[((size_t)bo * 128 + c) * 4096 + pos0 + rr] = val;  // coalesced over rr
  }
}

// ---------------------------------------------------------------------------
extern "C" void kernel_launch(void* const* d_in, const int* in_sizes, int n_in,
                              void* d_out, int out_size, void* d_ws, size_t ws_size,
                              hipStream_t stream) {
  (void)in_sizes; (void)n_in; (void)out_size; (void)ws_size;
  const float* x1   = (const float*)d_in[0];
  const float* x2   = (const float*)d_in[1];
  const float* Wt   = (const float*)d_in[2];
  const float* bt   = (const float*)d_in[3];
  const float* Wc   = (const float*)d_in[4];
  const float* bc   = (const float*)d_in[5];
  const float* bng  = (const float*)d_in[6];
  const float* bnb  = (const float*)d_in[7];
  const float* inw  = (const float*)d_in[8];
  const float* c1w  = (const float*)d_in[9];
  const float* c1b  = (const float*)d_in[10];
  const float* xpw  = (const float*)d_in[11];
  const float* dtw  = (const float*)d_in[12];
  const float* dtb  = (const float*)d_in[13];
  const float* alog = (const float*)d_in[14];
  const float* Dp   = (const float*)d_in[15];
  const float* opw  = (const float*)d_in[16];
  const float* lng  = (const float*)d_in[17];
  const float* lnb  = (const float*)d_in[18];
  float* out = (float*)d_out;
  float* w = (float*)d_ws;

  float* xcat  = w + OFF_XCAT;
  float* seq   = w + OFF_SEQ;
  float* xz    = w + OFF_XZ;
  float* xc    = w + OFF_XC;
  float* dbl   = w + OFF_DBL;
  float* dt    = w + OFF_DT;
  float* stats = w + OFF_STAT;
  float* x1cl  = w + OFF_X1CL;
  float* wtk   = w + OFF_WTK;
  float* wck   = w + OFF_WCK;
  float* yssm  = w + OFF_YSSM;

  // packing (layouts with contiguous K for WMMA operands)
  k_pack_x2<<<16384, 256, 0, stream>>>(x2, xcat);
  k_pack_x1<<<8192, 256, 0, stream>>>(x1, x1cl);
  k_pack_wt<<<512, 256, 0, stream>>>(Wt, wtk);
  k_pack_wc<<<1152, 256, 0, stream>>>(Wc, wck);
  // 1: conv_transpose -> xcat[...,128:256]
  k_upsample_wmma<<<dim3(512, 8, 4), 32, 0, stream>>>(x1cl, wtk, bt, xcat);
  // 2: conv3x3 (LDS halo) -> seq (pre-BN)
  k_conv3x3_wmma<<<2048, 128, 0, stream>>>(xcat, wck, bc, seq);
  // 3-4: batchnorm + relu
  hipMemsetAsync(stats, 0, 256 * sizeof(float), stream);
  k_bnstats<<<128, 128, 0, stream>>>(seq, stats);
  k_bnrelu<<<16384, 256, 0, stream>>>(seq, stats, bng, bnb);
  // 5: in_proj (TDM-staged A tile)
  k_inproj_wmma<<<512, 256, 0, stream>>>(seq, inw, xz);
  // 6: depthwise conv1d + silu
  k_conv1d_silu<<<32768, 256, 0, stream>>>(xz, c1w, c1b, xc);
  // 7: x_proj (N padded 40->48)
  k_xproj_wmma<<<dim3(2048, 3), 32, 0, stream>>>(xc, xpw, dbl);
  // 8: dt_proj + softplus
  k_dtproj_wmma<<<dim3(2048, 16), 32, 0, stream>>>(dbl, dtw, dtb, dt);
  // 9: selective scan (latency-critical sequential stage)
  k_scan<<<dim3(8, 4), 1024, 0, stream>>>(dt, dbl, xc, alog, Dp, yssm);
  // 10: gated out_proj + layernorm + NCHW store
  k_outproj_ln<<<2048, 128, 0, stream>>>(yssm, xz, opw, lng, lnb, out);
}